// CustomModel_Group_EB_MLP_Model_3753801417087
// MI455X (gfx1250) — compile-verified
//
#include <hip/hip_runtime.h>
#include <hip/hip_bf16.h>

typedef float v2f __attribute__((ext_vector_type(2)));
typedef float v8f __attribute__((ext_vector_type(8)));

// Workspace layout (floats):
//   ws[0   .. 255] : Wf zero-padded to [16][16]  (rows n>=3, cols k>=13 are 0)
//   ws[256 .. 271] : bf zero-padded to [16]      (entries n>=3 are 0)
#define WS_BIAS 256

// ---------------------------------------------------------------------------
// Kernel 1: collapse the activation-free 3-layer MLP into one affine map.
//   Wf[3][13] = w2(3x6) @ w1(6x12) @ w0(12x13)
//   bf[3]     = w2 @ (w1 @ b0 + b1) + b2
// 256 threads: one thread per padded Wf element (recomputes the 72-FMA dot
// product instead of staging W10 -> tiny code, fully parallel, writes the
// zero padding so the WMMA kernel needs no predicated loads).
// ---------------------------------------------------------------------------
__global__ void fuse_mlp_weights(const float* __restrict__ w0, const float* __restrict__ b0,
                                 const float* __restrict__ w1, const float* __restrict__ b1,
                                 const float* __restrict__ w2, const float* __restrict__ b2,
                                 float* __restrict__ ws) {
    const int tid = threadIdx.x;            // 0..255
    const int n = tid >> 4;                 // padded output row   0..15
    const int k = tid & 15;                 // padded K column     0..15

    float v = 0.0f;
    if (n < 3 && k < 13) {
        // Wf[n][k] = sum_j w2[n][j] * (sum_i w1[j][i] * w0[i][k])
        for (int j = 0; j < 6; ++j) {
            float w10 = 0.0f;
            for (int i = 0; i < 12; ++i) w10 += w1[j * 12 + i] * w0[i * 13 + k];
            v += w2[n * 6 + j] * w10;
        }
    }
    ws[n * 16 + k] = v;

    if (tid < 16) {
        float bias = 0.0f;
        if (tid < 3) {
            bias = b2[tid];
            for (int j = 0; j < 6; ++j) {
                float b10 = b1[j];
                for (int i = 0; i < 12; ++i) b10 += w1[j * 12 + i] * b0[i];
                bias += w2[tid * 6 + j] * b10;
            }
        }
        ws[WS_BIAS + tid] = bias;
    }
}

// ---------------------------------------------------------------------------
// Kernel 2: EmbeddingBag(mean). One wave32 per bag; lanes stride the bag's
// indices (<=2 independent random 12B gathers per lane -> deep memory-level
// parallelism across 16384 waves), butterfly shfl_xor reduction, lanes 0..8
// write the 3 replicated mean columns (out cols 0..8).
// ---------------------------------------------------------------------------
__global__ void eb_mean_kernel(const int* __restrict__ eb_input,
                               const int* __restrict__ eb_offset,
                               const float* __restrict__ emb,
                               float* __restrict__ out, int nbags, int ntotal) {
    const int wave = (int)((blockIdx.x * blockDim.x + threadIdx.x) >> 5);
    const int lane = (int)(threadIdx.x & 31);
    if (wave >= nbags) return;                       // uniform per wave

    const int start = eb_offset[wave];               // wave-uniform -> scalarized
    const int end   = (wave + 1 < nbags) ? eb_offset[wave + 1] : ntotal;
    const int cnt   = end - start;

    float sx = 0.0f, sy = 0.0f, sz = 0.0f;
    for (int i = lane; i < cnt; i += 32) {
        const int idx = eb_input[start + i];
        const float* r = emb + (long long)idx * 3;   // random 12B row gather
        sx += r[0];
        sy += r[1];
        sz += r[2];
    }
    // wave32 butterfly reduction: every lane ends with the full bag sum
    #pragma unroll
    for (int m = 16; m >= 1; m >>= 1) {
        sx += __shfl_xor(sx, m, 32);
        sy += __shfl_xor(sy, m, 32);
        sz += __shfl_xor(sz, m, 32);
    }
    const float inv = 1.0f / (float)cnt;
    if (lane < 9) {
        const int c = lane - (lane >= 6 ? 6 : (lane >= 3 ? 3 : 0)); // lane % 3
        const float v = (c == 0) ? sx : (c == 1) ? sy : sz;
        out[(long long)wave * 12 + lane] = v * inv;
    }
}

// ---------------------------------------------------------------------------
// Kernel 3: fused MLP as WMMA GEMM: out[:,9:12] = X(16384x13) @ Wf^T + bf.
// One wave per 16-row tile; K padded 13->16 -> 4x v_wmma_f32_16x16x4_f32
// accumulating into an 8-VGPR f32 C tile. Exact f32 arithmetic (no downcast).
//
// Fragment layouts per CDNA5 ISA 7.12.2:
//   A 16x4 f32 : lanes 0-15 {v0=K0,v1=K1}, lanes 16-31 {v0=K2,v1=K3}, M=lane%16
//   B 4x16 f32 : v0={K0 (lanes 0-15), K2 (lanes 16-31)}, v1={K1,K3}, N=lane%16
//   C/D 16x16  : VGPR v: lanes 0-15 -> M=v, lanes 16-31 -> M=v+8, N=lane%16
//
// All fragment loads are UNCONDITIONAL (ws is zero-padded; the single padded
// A element is a cndmask select on an always-valid load) -> no EXEC-mask
// branching before the WMMAs. Grid exactly covers B so EXEC is all ones.
// ---------------------------------------------------------------------------
__global__ void mlp_wmma_kernel(const float* __restrict__ x,
                                const float* __restrict__ ws,
                                float* __restrict__ out, int nrows) {
    const int lane   = (int)(threadIdx.x & 31);
    const int wave   = (int)((blockIdx.x * blockDim.x + threadIdx.x) >> 5);
    const int m0     = wave * 16;
    if (m0 >= nrows) return;                         // uniform; never taken
    const int half   = lane >> 4;                    // 0: lanes 0-15, 1: 16-31
    const int lane15 = lane & 15;
    const float* xr  = x + (long long)(m0 + lane15) * 13;

    // B fragments: lane's column of the zero-padded Wf^T tile.
    // ka = 4t + 2*half is always even and 8B-aligned in ws -> float2 loads.
    const float2* wrow = (const float2*)(ws + lane15 * 16);

    v8f c = {};
    #pragma unroll
    for (int t = 0; t < 4; ++t) {
        const int ka = 4 * t + 2 * half;             // K index of .x element
        v2f a;
        if (t < 3) {                                 // ka+1 <= 11 < 13: no pad
            a.x = xr[ka];
            a.y = xr[ka + 1];
        } else {                                     // only K=12 is live
            const float last = xr[12];               // always-valid address
            a.x = half ? 0.0f : last;                // v_cndmask, no branch
            a.y = 0.0f;
        }
        const float2 bw = wrow[2 * t + half];        // {Wf^T[ka][n], Wf^T[ka+1][n]}
        v2f b;
        b.x = bw.x;
        b.y = bw.y;
        c = __builtin_amdgcn_wmma_f32_16x16x4_f32(
                /*neg_a=*/false, a, /*neg_b=*/false, b,
                /*c_mod=*/(short)0, c, /*reuse_a=*/false, /*reuse_b=*/false);
    }

    const float bias = ws[WS_BIAS + lane15];         // unconditional (padded)
    if (lane15 < 3) {                                // only stores diverge
        #pragma unroll
        for (int v = 0; v < 8; ++v) {
            const int row = m0 + v + 8 * half;
            out[(long long)row * 12 + 9 + lane15] = c[v] + bias;
        }
    }
}

extern "C" void kernel_launch(void* const* d_in, const int* in_sizes, int n_in,
                              void* d_out, int out_size, void* d_ws, size_t ws_size,
                              hipStream_t stream) {
    const int*   eb_input  = (const int*)  d_in[0];
    const int*   eb_offset = (const int*)  d_in[1];
    const float* mlp_input = (const float*)d_in[2];
    const float* emb       = (const float*)d_in[3];
    const float* w0        = (const float*)d_in[4];
    const float* b0        = (const float*)d_in[5];
    const float* w1        = (const float*)d_in[6];
    const float* b1        = (const float*)d_in[7];
    const float* w2        = (const float*)d_in[8];
    const float* b2        = (const float*)d_in[9];
    float*       out       = (float*)d_out;
    float*       ws        = (float*)d_ws;

    const int ntotal = in_sizes[0];          // N = 819200
    const int nbags  = in_sizes[1];          // B = 16384

    // 1) collapse the linear MLP into padded Wf[16][16] + bf[16] in workspace
    fuse_mlp_weights<<<1, 256, 0, stream>>>(w0, b0, w1, b1, w2, b2, ws);

    // 2) embedding-bag mean: one wave32 per bag
    {
        const int threads = 256;                       // 8 waves / block
        const int blocks  = (nbags * 32 + threads - 1) / threads;
        eb_mean_kernel<<<blocks, threads, 0, stream>>>(eb_input, eb_offset, emb,
                                                       out, nbags, ntotal);
    }

    // 3) WMMA MLP: one wave per 16 rows (grid exactly covers B -> EXEC all 1s)
    {
        const int threads = 256;                       // 8 waves / block
        const int waves   = nbags / 16;                // 1024 tiles
        const int blocks  = (waves * 32 + threads - 1) / threads;
        mlp_wmma_kernel<<<blocks, threads, 0, stream>>>(mlp_input, ws, out, nbags);
    }
}